// LightGCL_57251914055926
// MI455X (gfx1250) — compile-verified
//
#include <hip/hip_runtime.h>
#include <cstdint>

#define N_USERS   100000
#define N_ITEMS   50000
#define DIM       64
#define NLAYER    2
#define RANK      32
#define NEDGE     1000000
#define BATCH     4096
#define KNEG      4
#define KEEP      0.8f
#define TEMP      0.2f
#define LAMBDA1   0.2f
#define ALPHA     1.0f

typedef __attribute__((ext_vector_type(2))) float v2f;
typedef __attribute__((ext_vector_type(8))) float v8f;

// ---------------------------------------------------------------- utilities
__global__ void k_zero(float* __restrict__ p, long n) {
    long t = (long)blockIdx.x * blockDim.x + threadIdx.x;
    if (t < n) p[t] = 0.0f;
}

__global__ void k_init(float* __restrict__ cur, float* __restrict__ sum,
                       const float* __restrict__ src, long n) {
    long t = (long)blockIdx.x * blockDim.x + threadIdx.x;
    if (t < n) { float v = src[t]; cur[t] = v; sum[t] = v; }
}

__global__ void k_leaky(float* __restrict__ p, long n) {
    long t = (long)blockIdx.x * blockDim.x + threadIdx.x;
    if (t < n) { float v = p[t]; p[t] = (v >= 0.0f) ? v : 0.5f * v; }
}

// cur += z; sum += cur
__global__ void k_update(float* __restrict__ cur, float* __restrict__ sum,
                         const float* __restrict__ z, long n) {
    long t = (long)blockIdx.x * blockDim.x + threadIdx.x;
    if (t < n) { float c = cur[t] + z[t]; cur[t] = c; sum[t] += c; }
}

// ------------------------------------------------------- SpMM (scatter-add)
// one thread per (edge, 4-dim chunk): b128 gather + 4 f32 atomics.
__global__ void k_spmm(const float* __restrict__ src_emb,
                       const float* __restrict__ vals,
                       const unsigned char* __restrict__ drop,
                       const int* __restrict__ rows,
                       const int* __restrict__ cols,
                       float* __restrict__ dst, long nedge) {
    long t = (long)blockIdx.x * blockDim.x + threadIdx.x;
    long e = t >> 4;
    int  q = (int)(t & 15) * 4;
    if (e >= nedge) return;
    if (!drop[e]) return;
    float v = vals[e] * (1.0f / KEEP);
    const float4 s = *(const float4*)&src_emb[(long)cols[e] * DIM + q];
    float* d = &dst[(long)rows[e] * DIM + q];
    atomicAdd(d + 0, v * s.x);
    atomicAdd(d + 1, v * s.y);
    atomicAdd(d + 2, v * s.z);
    atomicAdd(d + 3, v * s.w);
}

// ------------------------------------------- M = Wt[RANK,N] @ Emb[N,DIM]
// 64 threads/block (thread owns d), grid-strided chunks over N, atomic finish
__global__ void k_proj(const float* __restrict__ Wt, const float* __restrict__ Emb,
                       float* __restrict__ M, int N) {
    int d = threadIdx.x;
    float acc[RANK];
#pragma unroll
    for (int r = 0; r < RANK; ++r) acc[r] = 0.0f;
    int per = (N + gridDim.x - 1) / gridDim.x;
    int n0 = blockIdx.x * per;
    int n1 = n0 + per; if (n1 > N) n1 = N;
    for (int n = n0; n < n1; ++n) {
        float ev = Emb[(long)n * DIM + d];
#pragma unroll
        for (int r = 0; r < RANK; ++r) acc[r] += Wt[(long)r * N + n] * ev;
    }
#pragma unroll
    for (int r = 0; r < RANK; ++r) atomicAdd(&M[r * DIM + d], acc[r]);
}

// ------------------------------------- per-sample gnn / hyp feature build
__device__ inline float blocksum64(float v, float* sh, int d) {
    sh[d] = v; __syncthreads();
    for (int s = 32; s > 0; s >>= 1) { if (d < s) sh[d] += sh[d + s]; __syncthreads(); }
    float r = sh[0]; __syncthreads();
    return r;
}

// gnn[b] = l2n(Z[ids[b]]);  hyp[b] = l2n(act(mul_s[ids[b]] @ M)) @ Wl
__global__ void k_feat(const int* __restrict__ ids,
                       const float* __restrict__ Z,
                       const float* __restrict__ mul_s,
                       const float* __restrict__ M,
                       const float* __restrict__ Wl,
                       float* __restrict__ gnn, float* __restrict__ hyp) {
    __shared__ float sred[DIM];
    __shared__ float gsh[DIM];
    __shared__ float us[RANK];
    int b = blockIdx.x;
    int d = threadIdx.x;                 // 64 threads
    int u = ids[b];
    if (d < RANK) us[d] = mul_s[(long)u * RANK + d];
    __syncthreads();
    // g = act(us @ M)[d]
    float g = 0.0f;
#pragma unroll
    for (int r = 0; r < RANK; ++r) g += us[r] * M[r * DIM + d];
    g = (g >= 0.0f) ? g : 0.5f * g;
    // l2-normalize g
    float n2 = blocksum64(g * g, sred, d);
    float gn = g / fmaxf(sqrtf(n2), 1e-12f);
    gsh[d] = gn; __syncthreads();
    // hyp[d] = sum_dd gn[dd] * Wl[dd][d]
    float h = 0.0f;
    for (int dd = 0; dd < DIM; ++dd) h += gsh[dd] * Wl[dd * DIM + d];
    hyp[(long)b * DIM + d] = h;
    // gnn = l2n(Z[u])
    float z = Z[(long)u * DIM + d];
    float zn2 = blocksum64(z * z, sred, d);
    gnn[(long)b * DIM + d] = z / fmaxf(sqrtf(zn2), 1e-12f);
}

// ---------------------------------------------- neg_s via fp32 WMMA tiles
// 256-thread block computes a 64x64 tile of exp(gnn @ hyp^T / T) row-sums.
// A/B strips staged to LDS with async b128 loads (ASYNCcnt), 8 waves x two
// 16x16 WMMA sub-tiles each (K=64 = 16 chained 16x16x4 f32 WMMAs).
#define TI 64
#define TJ 64
__global__ void __launch_bounds__(256) k_neg(const float* __restrict__ gnn,
                                             const float* __restrict__ hyp,
                                             float* __restrict__ neg_s) {
    __shared__ __align__(16) float sA[TI * DIM];   // 16 KB gnn strip
    __shared__ __align__(16) float sB[TJ * DIM];   // 16 KB hyp strip
    const int bi = blockIdx.x / (BATCH / TJ);
    const int bj = blockIdx.x % (BATCH / TJ);
    const int t  = threadIdx.x;

    const float* gA = gnn + (long)bi * TI * DIM;
    const float* gB = hyp + (long)bj * TJ * DIM;
#pragma unroll
    for (int c = 0; c < (TI * DIM) / (256 * 4); ++c) {   // 4 iters, b128 each
        int idx4 = (t + 256 * c) * 4;                    // float index
        unsigned      ldsA = (unsigned)(size_t)&sA[idx4];  // flat->LDS: addr[31:0]
        unsigned      ldsB = (unsigned)(size_t)&sB[idx4];
        unsigned long gaA  = (unsigned long)(const void*)(gA + idx4);
        unsigned long gaB  = (unsigned long)(const void*)(gB + idx4);
        asm volatile("global_load_async_to_lds_b128 %0, %1, off"
                     :: "v"(ldsA), "v"(gaA) : "memory");
        asm volatile("global_load_async_to_lds_b128 %0, %1, off"
                     :: "v"(ldsB), "v"(gaB) : "memory");
    }
    asm volatile("s_wait_asynccnt 0" ::: "memory");
    __syncthreads();

    const int wave = t >> 5;
    const int lane = t & 31;
    const int m    = lane & 15;
    const int kb   = (lane >> 4) << 1;   // K pair this half-wave owns
    const float invT = 1.0f / TEMP;
#pragma unroll
    for (int s = 0; s < 2; ++s) {
        const int st  = wave + 8 * s;    // sub-tile 0..15 in 4x4 grid
        const int tis = st >> 2, tjs = st & 3;
        const float* arow = &sA[(tis * 16 + m) * DIM + kb];
        const float* brow = &sB[(tjs * 16 + m) * DIM + kb];
        v8f c = {};
#pragma unroll
        for (int kk = 0; kk < 16; ++kk) {
            v2f a; a.x = arow[kk * 4]; a.y = arow[kk * 4 + 1];
            v2f b; b.x = brow[kk * 4]; b.y = brow[kk * 4 + 1];
            c = __builtin_amdgcn_wmma_f32_16x16x4_f32(false, a, false, b,
                                                      (short)0, c, false, false);
        }
        // C layout: VGPR v -> row v (lanes 0-15) / v+8 (lanes 16-31), col = lane&15
#pragma unroll
        for (int v = 0; v < 8; ++v) {
            float e = expf(c[v] * invT);
            for (int off = 1; off < 16; off <<= 1) e += __shfl_xor(e, off, 32);
            if (m == 0) {
                int row = bi * TI + tis * 16 + v + ((lane >> 4) << 3);
                atomicAdd(&neg_s[row], e);
            }
        }
    }
}

// -------------------------------------------------- contrastive loss terms
__global__ void k_loss_s(const float* __restrict__ gnn, const float* __restrict__ hyp,
                         const float* __restrict__ neg_s,
                         const unsigned char* __restrict__ mask,
                         float* __restrict__ acc) {
    int b = blockIdx.x * blockDim.x + threadIdx.x;
    float v = 0.0f;
    if (b < BATCH) {
        float dot = 0.0f;
        for (int d = 0; d < DIM; ++d) dot += gnn[(long)b * DIM + d] * hyp[(long)b * DIM + d];
        float pos  = expf(dot * (1.0f / TEMP));
        float term = -logf(pos / (neg_s[b] + 1e-8f) + 1e-8f);
        v = mask[b] ? term : 0.0f;
    }
    for (int off = 1; off < 32; off <<= 1) v += __shfl_xor(v, off, 32);
    if ((threadIdx.x & 31) == 0) atomicAdd(acc, v);
}

// --------------------------------------------------------------- BPR loss
// one wave per sample; thread owns dims (lane, lane+32)
__global__ void __launch_bounds__(32) k_bpr(const float* __restrict__ Eu,
                                            const float* __restrict__ Ei,
                                            const int* __restrict__ uids,
                                            const int* __restrict__ pos,
                                            const int* __restrict__ neg,
                                            const int* __restrict__ cat_of_user,
                                            const float* __restrict__ noise,
                                            const float* __restrict__ eps,
                                            float* __restrict__ acc) {
    int b = blockIdx.x, lane = threadIdx.x;
    int u   = uids[b];
    int cat = cat_of_user[u];
    float ec = eps[cat];
    int d0 = lane, d1 = lane + 32;
    float nA = noise[(long)b * DIM + d0], nB = noise[(long)b * DIM + d1];
    float nn2 = nA * nA + nB * nB;
    for (int off = 1; off < 32; off <<= 1) nn2 += __shfl_xor(nn2, off, 32);
    float inv = 1.0f / fmaxf(sqrtf(nn2), 1e-12f);
    float e0 = Eu[(long)u * DIM + d0], e1 = Eu[(long)u * DIM + d1];
    float s0 = (e0 > 0.0f) ? 1.0f : ((e0 < 0.0f) ? -1.0f : 0.0f);
    float s1 = (e1 > 0.0f) ? 1.0f : ((e1 < 0.0f) ? -1.0f : 0.0f);
    float u0 = e0 + s0 * nA * inv * ec;
    float u1 = e1 + s1 * nB * inv * ec;
    float per_u = 0.0f;
#pragma unroll
    for (int k = 0; k < KNEG; ++k) {
        int p  = pos[b * KNEG + k];
        int ng = neg[b * KNEG + k];
        float ps = u0 * Ei[(long)p * DIM + d0] + u1 * Ei[(long)p * DIM + d1];
        float ns = u0 * Ei[(long)ng * DIM + d0] + u1 * Ei[(long)ng * DIM + d1];
        for (int off = 1; off < 32; off <<= 1) {
            ps += __shfl_xor(ps, off, 32);
            ns += __shfl_xor(ns, off, 32);
        }
        per_u += fmaxf(0.0f, 1.0f - ps + ns);
    }
    if (lane == 0) {
        atomicAdd(&acc[1], per_u);        // loss_r total
        atomicAdd(&acc[2 + cat], per_u);  // seg[cat]
        atomicAdd(&acc[5 + cat], 1.0f);   // cnt[cat]
    }
}

__global__ void k_final(const float* __restrict__ acc, float* __restrict__ out) {
    if (threadIdx.x != 0 || blockIdx.x != 0) return;
    float loss_s = acc[0];
    float loss_r = acc[1] / (float)BATCH;
    float lcat = 0.0f;
#pragma unroll
    for (int c = 0; c < 3; ++c) {
        float cnt = acc[5 + c];
        if (cnt > 0.0f) lcat += acc[2 + c] / fmaxf(cnt, 1.0f);
    }
    out[0] = ALPHA * lcat + LAMBDA1 * loss_s;
    out[1] = loss_r;
    out[2] = loss_s;
}

// ---------------------------------------------------------------- launcher
extern "C" void kernel_launch(void* const* d_in, const int* in_sizes, int n_in,
                              void* d_out, int out_size, void* d_ws, size_t ws_size,
                              hipStream_t stream) {
    const float* E_u_0   = (const float*)d_in[0];
    const float* E_i_0   = (const float*)d_in[1];
    const float* Ws      = (const float*)d_in[2];
    const float* adjval  = (const float*)d_in[3];
    const float* u_mul_s = (const float*)d_in[4];
    const float* v_mul_s = (const float*)d_in[5];
    const float* ut      = (const float*)d_in[6];
    const float* vt      = (const float*)d_in[7];
    const float* noise_u = (const float*)d_in[8];
    const float* eps     = (const float*)d_in[9];
    const int*   adjrow  = (const int*)d_in[10];
    const int*   adjcol  = (const int*)d_in[11];
    const unsigned char* drop_u = (const unsigned char*)d_in[12];
    const unsigned char* drop_i = (const unsigned char*)d_in[13];
    const unsigned char* u_mask = (const unsigned char*)d_in[14];
    const unsigned char* i_mask = (const unsigned char*)d_in[15];
    const int*   uids    = (const int*)d_in[16];
    const int*   iids    = (const int*)d_in[17];
    const int*   pos     = (const int*)d_in[18];
    const int*   neg     = (const int*)d_in[19];
    const int*   acat    = (const int*)d_in[20];
    float* out = (float*)d_out;

    // ---- workspace carve-up (floats)
    float* ws = (float*)d_ws;
    size_t o = 0;
    const long NU64 = (long)N_USERS * DIM, NI64 = (long)N_ITEMS * DIM;
    float* Eu_cur = ws + o; o += NU64;
    float* Ei_cur = ws + o; o += NI64;
    float* Eu_sum = ws + o; o += NU64;
    float* Ei_sum = ws + o; o += NI64;
    float* Zu     = ws + o; o += NU64;
    float* Zi     = ws + o; o += NI64;
    float* M_u    = ws + o; o += RANK * DIM;
    float* M_i    = ws + o; o += RANK * DIM;
    float* gnn_u  = ws + o; o += (long)BATCH * DIM;
    float* hyp_u  = ws + o; o += (long)BATCH * DIM;
    float* gnn_i  = ws + o; o += (long)BATCH * DIM;
    float* hyp_i  = ws + o; o += (long)BATCH * DIM;
    float* neg_su = ws + o; o += BATCH;
    float* neg_si = ws + o; o += BATCH;
    float* acc    = ws + o; o += 8;   // [0]=loss_s [1]=loss_r_sum [2..4]=seg [5..7]=cnt

    auto gblk = [](long n) { return (unsigned)((n + 255) / 256); };

    // init: cur = sum = E0 ; zero accumulators
    k_init<<<gblk(NU64), 256, 0, stream>>>(Eu_cur, Eu_sum, E_u_0, NU64);
    k_init<<<gblk(NI64), 256, 0, stream>>>(Ei_cur, Ei_sum, E_i_0, NI64);
    k_zero<<<1, 8, 0, stream>>>(acc, 8);

    const long spmm_threads = (long)NEDGE * 16;                 // 4 dims / thread
    const unsigned NEGBLKS = (BATCH / TI) * (BATCH / TJ);       // 4096

    for (int layer = 0; layer < NLAYER; ++layer) {
        const float* Wl = Ws + (long)layer * DIM * DIM;
        // zero per-layer scratch
        k_zero<<<gblk(NU64), 256, 0, stream>>>(Zu, NU64);
        k_zero<<<gblk(NI64), 256, 0, stream>>>(Zi, NI64);
        k_zero<<<gblk(2 * RANK * DIM), 256, 0, stream>>>(M_u, 2 * RANK * DIM); // M_u+M_i contiguous
        k_zero<<<gblk(2 * BATCH), 256, 0, stream>>>(neg_su, 2 * BATCH);        // neg_su+neg_si contiguous

        // SpMM both sides (pre-activation scatter)
        k_spmm<<<gblk(spmm_threads), 256, 0, stream>>>(Ei_cur, adjval,
            drop_u + (size_t)layer * NEDGE, adjrow, adjcol, Zu, NEDGE);
        k_spmm<<<gblk(spmm_threads), 256, 0, stream>>>(Eu_cur, adjval,
            drop_i + (size_t)layer * NEDGE, adjcol, adjrow, Zi, NEDGE);
        k_leaky<<<gblk(NU64), 256, 0, stream>>>(Zu, NU64);
        k_leaky<<<gblk(NI64), 256, 0, stream>>>(Zi, NI64);

        // rank-space projections: M_u = vt@Ei_cur, M_i = ut@Eu_cur (pre-update E)
        k_proj<<<256, DIM, 0, stream>>>(vt, Ei_cur, M_u, N_ITEMS);
        k_proj<<<256, DIM, 0, stream>>>(ut, Eu_cur, M_i, N_USERS);

        // sampled gnn / hyp features
        k_feat<<<BATCH, DIM, 0, stream>>>(uids, Zu, u_mul_s, M_u, Wl, gnn_u, hyp_u);
        k_feat<<<BATCH, DIM, 0, stream>>>(iids, Zi, v_mul_s, M_i, Wl, gnn_i, hyp_i);

        // B x B exp row-sums: async-LDS staged fp32 WMMA
        k_neg<<<NEGBLKS, 256, 0, stream>>>(gnn_u, hyp_u, neg_su);
        k_neg<<<NEGBLKS, 256, 0, stream>>>(gnn_i, hyp_i, neg_si);

        // contrastive loss accumulation
        k_loss_s<<<(BATCH + 127) / 128, 128, 0, stream>>>(gnn_u, hyp_u, neg_su,
            u_mask + (size_t)layer * BATCH, acc);
        k_loss_s<<<(BATCH + 127) / 128, 128, 0, stream>>>(gnn_i, hyp_i, neg_si,
            i_mask + (size_t)layer * BATCH, acc);

        // E_cur += Z ; E_sum += E_cur
        k_update<<<gblk(NU64), 256, 0, stream>>>(Eu_cur, Eu_sum, Zu, NU64);
        k_update<<<gblk(NI64), 256, 0, stream>>>(Ei_cur, Ei_sum, Zi, NI64);
    }

    // BPR with per-category noise + category-mean loss
    k_bpr<<<BATCH, 32, 0, stream>>>(Eu_sum, Ei_sum, uids, pos, neg, acat,
                                    noise_u, eps, acc);
    k_final<<<1, 1, 0, stream>>>(acc, out);
}